// DeformableConv2d_6811818132510
// MI455X (gfx1250) — compile-verified
//
#include <hip/hip_runtime.h>
#include <math.h>

// ---------------------------------------------------------------------------
// Deformable Conv2d (DCNv2) for MI455X / gfx1250, fp32 via V_WMMA_F32_16X16X4_F32
//
//   x [8,64,128,128] f32 -> out [8,64,128,128] f32
//
// ~14.4 GFLOP fp32, ~100 MB HBM traffic; x (34 MB) is L2-resident (192 MB L2)
// so bilinear gathers are L2 hits -> fp32-WMMA compute bound.
// Kernel B is software-pipelined: next K-chunk's 16 gathers/pair are issued
// into registers, the current chunk's 18 WMMAs hide the latency, then the
// sampled tile is stored into the alternate LDS buffer (1 barrier / chunk).
// ---------------------------------------------------------------------------

typedef __attribute__((ext_vector_type(2))) float v2f;
typedef __attribute__((ext_vector_type(8))) float v8f;

#define CIN      64
#define COUT     64
#define KK       9          // 3x3 taps
#define H        128
#define W        128
#define HW       (H * W)
#define NB       8
#define PIX_TILE 64         // pixels per block (half an image row -> same oy)
#define KCH      36         // K-chunk: 4 channels * 9 taps (channel aligned)
#define NCHUNK   16         // 576 / 36
#define NPAIR    3          // 9*64 = 576 pairs = 2*256 + 64

// ---------------------------------------------------------------------------
// Kernel A: offset + modulation convs as WMMA GEMM  W[32(27 used),576] x col
// ---------------------------------------------------------------------------
__global__ __launch_bounds__(256)
void offmask_kernel(const float* __restrict__ x,
                    const float* __restrict__ offset_w,
                    const float* __restrict__ offset_b,
                    const float* __restrict__ mod_w,
                    const float* __restrict__ mod_b,
                    float* __restrict__ off_ws,    // [B][18][HW]
                    float* __restrict__ mask_ws)   // [B][ 9][HW]
{
    __shared__ float A_lds[32][KCH + 1];       // +1 pad: conflict-free frag reads
    __shared__ float B_lds[KCH][PIX_TILE];

    const int tid  = threadIdx.x;
    const int lane = tid & 31;
    const int wv   = tid >> 5;                 // wave 0..7
    const int b    = blockIdx.x >> 8;          // 256 pixel tiles per batch
    const int p0   = (blockIdx.x & 255) * PIX_TILE;
    const int oy   = p0 >> 7;                  // tile lies in one image row
    const int ox0  = p0 & 127;

    const float* xb = x + (size_t)b * CIN * HW;

    const int mt = wv >> 2;                    // 2 M-tiles (rows 0..31)
    const int nt = wv & 3;                     // 4 N-tiles (64 pixels)

    const int ml = mt * 16 + (lane & 15);
    const int nl = nt * 16 + (lane & 15);
    const int kh = (lane >> 4) << 1;           // 0 (lanes 0-15) or 2 (16-31)

    v8f acc = {};

    for (int cc = 0; cc < NCHUNK; ++cc) {
        // stage A: rows 0..17 = offset_w, 18..26 = mod_w, 27..31 = 0
        // 1152 entries = 4*256 + 128
#pragma unroll
        for (int it = 0; it < 5; ++it) {
            const int i = tid + it * 256;
            if (it < 4 || tid < 128) {
                const int m = i / KCH, t = i % KCH;
                const int kk = cc * KCH + t;   // kk = c*9 + tap
                float wval;
                if (m < 18)      wval = offset_w[m * 576 + kk];
                else if (m < 27) wval = mod_w[(m - 18) * 576 + kk];
                else             wval = 0.0f;
                A_lds[m][t] = wval;
            }
        }
        // stage B: plain im2col of x (zero padded); 2304 entries = 9*256 exact
#pragma unroll
        for (int it = 0; it < 9; ++it) {
            const int i = tid + it * 256;
            const int t = i >> 6, p = i & 63;
            const int c  = (cc * KCH + t) / KK;
            const int ks = t % KK;
            const int sy = oy + ks / 3 - 1;
            const int sx = ox0 + p + ks % 3 - 1;
            float v = 0.0f;
            if (sy >= 0 && sy < H && sx >= 0 && sx < W)
                v = xb[c * HW + sy * W + sx];
            B_lds[t][p] = v;
        }
        __syncthreads();

#pragma unroll
        for (int s = 0; s < KCH / 4; ++s) {
            const int kb = s * 4 + kh;
            v2f a, bf;
            a.x  = A_lds[ml][kb];     a.y  = A_lds[ml][kb + 1];
            bf.x = B_lds[kb][nl];     bf.y = B_lds[kb + 1][nl];
            acc = __builtin_amdgcn_wmma_f32_16x16x4_f32(
                      false, a, false, bf, (short)0, acc, false, false);
        }
        __syncthreads();
    }

    // D layout: vgpr r -> M = mt*16 + r (+8 for lanes 16..31), N = lane&15
    const int p = p0 + nt * 16 + (lane & 15);
#pragma unroll
    for (int r = 0; r < 8; ++r) {
        const int m = mt * 16 + r + ((lane >> 4) << 3);
        const float v = acc[r];
        if (m < 18) {
            off_ws[((size_t)b * 18 + m) * HW + p] = v + offset_b[m];
        } else if (m < 27) {
            const float z = v + mod_b[m - 18];
            mask_ws[((size_t)b * 9 + (m - 18)) * HW + p] =
                2.0f / (1.0f + __expf(-z));    // 2*sigmoid
        }
    }
}

// ---------------------------------------------------------------------------
// Kernel B: fused deformable sampling + WMMA GEMM, software pipelined with
// double-buffered LDS.   out[b,:,p] = reg_w[64,576] x S[576, pixels]
// ---------------------------------------------------------------------------
__global__ __launch_bounds__(256)
void dcn_gemm_kernel(const float* __restrict__ x,
                     const float* __restrict__ reg_w,
                     const float* __restrict__ off_ws,
                     const float* __restrict__ mask_ws,
                     float* __restrict__ out)
{
    __shared__ float A_lds[2][COUT][KCH + 1];  // 2 x 64 x 37 f32
    __shared__ float B_lds[2][KCH][PIX_TILE];  // 2 x 36 x 64 f32

    const int tid  = threadIdx.x;
    const int lane = tid & 31;
    const int wv   = tid >> 5;
    const int b    = blockIdx.x >> 8;
    const int p0   = (blockIdx.x & 255) * PIX_TILE;
    const int oy   = p0 >> 7;
    const int ox0  = p0 & 127;

    const float* xb = x + (size_t)b * CIN * HW;

    const int mt  = wv & 3;                    // 4 M-tiles (COUT=64)
    const int nt0 = (wv >> 2) * 2;             // pair of N-tiles per wave
    const int nt1 = nt0 + 1;

    const int ml = mt * 16 + (lane & 15);
    const int n0 = nt0 * 16 + (lane & 15);
    const int n1 = nt1 * 16 + (lane & 15);
    const int kh = (lane >> 4) << 1;

    // -------- bilinear precompute: indices + mask-folded corner weights ----
    // Pairs 0,1 exist for every thread; pair 2 only for tid < 64.
    int   gidx[NPAIR][4];
    float gwgt[NPAIR][4];
#pragma unroll
    for (int r = 0; r < NPAIR; ++r) {
        const int pr = tid + 256 * r;
        if (r < 2 || tid < 64) {
            const int ks = pr >> 6;
            const int p  = pr & 63;
            const int pg = p0 + p;
            const float dy = off_ws[((size_t)b * 18 + 2 * ks    ) * HW + pg];
            const float dx = off_ws[((size_t)b * 18 + 2 * ks + 1) * HW + pg];
            const float mk = mask_ws[((size_t)b * 9 + ks) * HW + pg];

            const float py = (float)(oy + ks / 3 - 1) + dy;
            const float px = (float)(ox0 + p + ks % 3 - 1) + dx;
            const float y0f = floorf(py), x0f = floorf(px);
            const float wy = py - y0f,  wx = px - x0f;
            const int y0 = (int)y0f, x0 = (int)x0f;
            const int y1 = y0 + 1,   x1 = x0 + 1;

            const float vy0 = (y0 >= 0 && y0 < H) ? 1.0f : 0.0f;
            const float vy1 = (y1 >= 0 && y1 < H) ? 1.0f : 0.0f;
            const float vx0 = (x0 >= 0 && x0 < W) ? 1.0f : 0.0f;
            const float vx1 = (x1 >= 0 && x1 < W) ? 1.0f : 0.0f;
            const int yc0 = y0 < 0 ? 0 : (y0 > H - 1 ? H - 1 : y0);
            const int yc1 = y1 < 0 ? 0 : (y1 > H - 1 ? H - 1 : y1);
            const int xc0 = x0 < 0 ? 0 : (x0 > W - 1 ? W - 1 : x0);
            const int xc1 = x1 < 0 ? 0 : (x1 > W - 1 ? W - 1 : x1);

            gwgt[r][0] = (1.0f - wy) * (1.0f - wx) * vy0 * vx0 * mk;
            gwgt[r][1] = (1.0f - wy) * wx          * vy0 * vx1 * mk;
            gwgt[r][2] = wy * (1.0f - wx)          * vy1 * vx0 * mk;
            gwgt[r][3] = wy * wx                   * vy1 * vx1 * mk;
            gidx[r][0] = yc0 * W + xc0;  gidx[r][1] = yc0 * W + xc1;
            gidx[r][2] = yc1 * W + xc0;  gidx[r][3] = yc1 * W + xc1;
        } else {
#pragma unroll
            for (int q = 0; q < 4; ++q) { gidx[r][q] = 0; gwgt[r][q] = 0.0f; }
        }
    }

    // A staging mapping (division-free): 4 threads per weight row, 9 each
    const int am  = tid >> 2;                  // 0..63
    const int at0 = (tid & 3) * 9;             // 0,9,18,27
    const float* wbase = reg_w + am * 576 + at0;

    float areg[9];                             // A prefetch registers
    float breg[NPAIR][16];                     // gather prefetch (4 ch x 4 corners)

    // ---- prologue: prefetch + store chunk 0 -------------------------------
#pragma unroll
    for (int j = 0; j < 9; ++j) areg[j] = wbase[j];
    {
        const float* xc = xb;                  // chunk 0 channels 0..3
#pragma unroll
        for (int r = 0; r < NPAIR; ++r) {
            if (r < 2 || tid < 64) {
#pragma unroll
                for (int cl = 0; cl < 4; ++cl) {
                    const float* xp = xc + cl * HW;
#pragma unroll
                    for (int q = 0; q < 4; ++q)
                        breg[r][cl * 4 + q] = xp[gidx[r][q]];
                }
            }
        }
    }
#pragma unroll
    for (int j = 0; j < 9; ++j) A_lds[0][am][at0 + j] = areg[j];
#pragma unroll
    for (int r = 0; r < NPAIR; ++r) {
        if (r < 2 || tid < 64) {
            const int pr = tid + 256 * r;
            const int ks = pr >> 6, p = pr & 63;
#pragma unroll
            for (int cl = 0; cl < 4; ++cl)
                B_lds[0][cl * KK + ks][p] =
                    gwgt[r][0] * breg[r][cl * 4 + 0] +
                    gwgt[r][1] * breg[r][cl * 4 + 1] +
                    gwgt[r][2] * breg[r][cl * 4 + 2] +
                    gwgt[r][3] * breg[r][cl * 4 + 3];
        }
    }
    __syncthreads();

    v8f acc0 = {}, acc1 = {};

    // ---- main pipelined loop: 1 barrier per chunk -------------------------
    for (int cc = 0; cc < NCHUNK; ++cc) {
        const int cur = cc & 1;

        // issue next chunk's global loads (in flight during the WMMAs)
        if (cc + 1 < NCHUNK) {
#pragma unroll
            for (int j = 0; j < 9; ++j) areg[j] = wbase[(cc + 1) * KCH + j];
            const float* xc = xb + (cc + 1) * 4 * HW;
#pragma unroll
            for (int r = 0; r < NPAIR; ++r) {
                if (r < 2 || tid < 64) {
#pragma unroll
                    for (int cl = 0; cl < 4; ++cl) {
                        const float* xp = xc + cl * HW;
#pragma unroll
                        for (int q = 0; q < 4; ++q)
                            breg[r][cl * 4 + q] = xp[gidx[r][q]];
                    }
                }
            }
        }

        // 18 WMMAs on the current buffer (hides the gather latency)
#pragma unroll
        for (int s = 0; s < KCH / 4; ++s) {
            const int kb = s * 4 + kh;
            v2f a, bA, bB;
            a.x  = A_lds[cur][ml][kb];      a.y  = A_lds[cur][ml][kb + 1];
            bA.x = B_lds[cur][kb][n0];      bA.y = B_lds[cur][kb + 1][n0];
            bB.x = B_lds[cur][kb][n1];      bB.y = B_lds[cur][kb + 1][n1];
            acc0 = __builtin_amdgcn_wmma_f32_16x16x4_f32(
                       false, a, false, bA, (short)0, acc0, false, false);
            acc1 = __builtin_amdgcn_wmma_f32_16x16x4_f32(
                       false, a, false, bB, (short)0, acc1, false, false);
        }

        // combine + store next chunk into the alternate buffer
        if (cc + 1 < NCHUNK) {
            const int nxt = cur ^ 1;
#pragma unroll
            for (int j = 0; j < 9; ++j) A_lds[nxt][am][at0 + j] = areg[j];
#pragma unroll
            for (int r = 0; r < NPAIR; ++r) {
                if (r < 2 || tid < 64) {
                    const int pr = tid + 256 * r;
                    const int ks = pr >> 6, p = pr & 63;
#pragma unroll
                    for (int cl = 0; cl < 4; ++cl)
                        B_lds[nxt][cl * KK + ks][p] =
                            gwgt[r][0] * breg[r][cl * 4 + 0] +
                            gwgt[r][1] * breg[r][cl * 4 + 1] +
                            gwgt[r][2] * breg[r][cl * 4 + 2] +
                            gwgt[r][3] * breg[r][cl * 4 + 3];
                }
            }
        }
        __syncthreads();
    }

    // ---- epilogue: store 64x64 output tile --------------------------------
#pragma unroll
    for (int r = 0; r < 8; ++r) {
        const int m = mt * 16 + r + ((lane >> 4) << 3);
        out[((size_t)b * COUT + m) * HW + p0 + n0] = acc0[r];
        out[((size_t)b * COUT + m) * HW + p0 + n1] = acc1[r];
    }
}

// ---------------------------------------------------------------------------
extern "C" void kernel_launch(void* const* d_in, const int* in_sizes, int n_in,
                              void* d_out, int out_size, void* d_ws, size_t ws_size,
                              hipStream_t stream) {
    const float* x        = (const float*)d_in[0];
    const float* offset_w = (const float*)d_in[1];
    const float* offset_b = (const float*)d_in[2];
    const float* mod_w    = (const float*)d_in[3];
    const float* mod_b    = (const float*)d_in[4];
    const float* reg_w    = (const float*)d_in[5];
    float* out = (float*)d_out;

    // workspace: off [8][18][HW] then mask [8][9][HW]  (~14.2 MB total)
    float* off_ws  = (float*)d_ws;
    float* mask_ws = off_ws + (size_t)NB * 18 * HW;

    const dim3 grid(NB * (HW / PIX_TILE));   // 8 * 256 = 2048 blocks
    const dim3 block(256);                   // 8 wave32

    offmask_kernel<<<grid, block, 0, stream>>>(x, offset_w, offset_b,
                                               mod_w, mod_b, off_ws, mask_ws);
    dcn_gemm_kernel<<<grid, block, 0, stream>>>(x, reg_w, off_ws, mask_ws, out);
}